// STBlock_82867099009444
// MI455X (gfx1250) — compile-verified
//
#include <hip/hip_runtime.h>
#include <math.h>

// ---------------------------------------------------------------------------
// Shapes: B=16, N=2048, T1=12, T2=64, K=3.
// Key observation: Ssc/S/_A_hat branch in the reference is dead code.
// Pipeline:
//   q,r (reduce over N) -> E (16x12x12 softmax block) -> X_hat = T0 [N x 192]
//   dis from A row sums
//   out  = T0 @ Wcheb0
//   T1   = M^T T0                       (WMMA f32 GEMM, 2048x2048x192)
//   out += T1 @ Wcheb1
//   T0   = 2*M^T T1 - T0                (WMMA f32 GEMM, alias-safe epilogue)
//   out  = relu(out + T0 @ Wcheb2 + bcheb)
// ---------------------------------------------------------------------------

typedef __attribute__((ext_vector_type(2))) float v2f;
typedef __attribute__((ext_vector_type(8))) float v8f;

#define NN   2048
#define T1D  12
#define T2D  64
#define BCOL 192          // B*T1 fused columns

// ---------------- q[b,t] = sum_n X[b,n,t]*U1[n];  r[b,s] = U3*sum_n U2[n]*X[b,n,s]
__global__ __launch_bounds__(256) void qr_kernel(const float* __restrict__ X,
                                                 const float* __restrict__ U1,
                                                 const float* __restrict__ U2,
                                                 const float* __restrict__ U3,
                                                 float* __restrict__ q,
                                                 float* __restrict__ r) {
    __shared__ float sq[T1D], sr[T1D];
    const int b = blockIdx.x, tid = threadIdx.x;
    if (tid < T1D) { sq[tid] = 0.f; sr[tid] = 0.f; }
    __syncthreads();
    float aq[T1D], ar[T1D];
#pragma unroll
    for (int t = 0; t < T1D; ++t) { aq[t] = 0.f; ar[t] = 0.f; }
    for (int n = tid; n < NN; n += 256) {
        const float u1 = U1[n], u2 = U2[n];
        const float* xp = X + ((size_t)b * NN + n) * T1D;
#pragma unroll
        for (int t = 0; t < T1D; ++t) { const float x = xp[t]; aq[t] += x * u1; ar[t] += x * u2; }
    }
#pragma unroll
    for (int t = 0; t < T1D; ++t) { atomicAdd(&sq[t], aq[t]); atomicAdd(&sr[t], ar[t]); }
    __syncthreads();
    if (tid < T1D) { q[b * T1D + tid] = sq[tid]; r[b * T1D + tid] = U3[0] * sr[tid]; }
}

// ---------------- E[b,t,u]: softmax_t(Esc) -> Ve @ . -> softmax_t
__global__ __launch_bounds__(160) void e_kernel(const float* __restrict__ q,
                                                const float* __restrict__ r,
                                                const float* __restrict__ be,
                                                const float* __restrict__ Ve,
                                                float* __restrict__ E) {
    const int b = blockIdx.x, tid = threadIdx.x;
    __shared__ float P[T1D][T1D], G[T1D][T1D];
    if (tid < 144) {
        const int t = tid / T1D, s = tid % T1D;
        P[t][s] = q[b * T1D + t] * r[b * T1D + s] + be[t * T1D + s];
    }
    __syncthreads();
    if (tid < T1D) {                       // softmax over t for column s
        const int s = tid; float mx = -INFINITY;
        for (int t = 0; t < T1D; ++t) mx = fmaxf(mx, P[t][s]);
        float sum = 0.f;
        for (int t = 0; t < T1D; ++t) { const float e = expf(P[t][s] - mx); P[t][s] = e; sum += e; }
        const float inv = 1.f / sum;
        for (int t = 0; t < T1D; ++t) P[t][s] *= inv;
    }
    __syncthreads();
    if (tid < 144) {
        const int t = tid / T1D, u = tid % T1D; float acc = 0.f;
        for (int s = 0; s < T1D; ++s) acc += Ve[t * T1D + s] * P[s][u];
        G[t][u] = acc;
    }
    __syncthreads();
    if (tid < T1D) {                       // softmax over t for column u
        const int u = tid; float mx = -INFINITY;
        for (int t = 0; t < T1D; ++t) mx = fmaxf(mx, G[t][u]);
        float sum = 0.f;
        for (int t = 0; t < T1D; ++t) { const float e = expf(G[t][u] - mx); G[t][u] = e; sum += e; }
        const float inv = 1.f / sum;
        for (int t = 0; t < T1D; ++t) G[t][u] *= inv;
    }
    __syncthreads();
    if (tid < 144) E[b * 144 + tid] = G[tid / T1D][tid % T1D];
}

// ---------------- T0[n, b*12+u] = sum_t X[b,n,t]*E[b,t,u]
__global__ __launch_bounds__(256) void xhat_kernel(const float* __restrict__ X,
                                                   const float* __restrict__ E,
                                                   float* __restrict__ T0) {
    const int i = blockIdx.x * 256 + threadIdx.x;      // [0, 2048*192)
    const int n = i / BCOL, col = i % BCOL;
    const int b = col / T1D, u = col % T1D;
    const float* xp = X + ((size_t)b * NN + n) * T1D;
    const float* ep = E + b * 144 + u;
    float acc = 0.f;
#pragma unroll
    for (int t = 0; t < T1D; ++t) acc += xp[t] * ep[t * T1D];
    T0[(size_t)n * BCOL + col] = acc;
}

// ---------------- dis[r] = deg>0 ? rsqrt(deg) : 0, deg = row sum of A
__global__ __launch_bounds__(256) void deg_kernel(const float* __restrict__ A,
                                                  float* __restrict__ dis) {
    const int row = blockIdx.x;
    __shared__ float s[256];
    float acc = 0.f;
    for (int c = threadIdx.x; c < NN; c += 256) acc += A[(size_t)row * NN + c];
    s[threadIdx.x] = acc;
    __syncthreads();
    for (int o = 128; o > 0; o >>= 1) {
        if (threadIdx.x < o) s[threadIdx.x] += s[threadIdx.x + o];
        __syncthreads();
    }
    if (threadIdx.x == 0) { const float d = s[0]; dis[row] = d > 0.f ? rsqrtf(d) : 0.f; }
}

// ---------------- Y[c,j] = alpha * sum_r (-dis[r]*A[r,c]*dis[c]) * H[r,j] + beta*Cin[c,j]
// 12 waves/block, wave j -> 16-col tile; block -> 16-row strip; K chunks of 64.
// Uses V_WMMA_F32_16X16X4_F32 (fp32 in, fp32 accum — matches reference precision).
#define TK 64
__global__ __launch_bounds__(384) void lx_gemm(const float* __restrict__ A,
                                               const float* __restrict__ dis,
                                               const float* __restrict__ H,
                                               const float* Cin,   // may alias Y (1:1 element map)
                                               float* Y,
                                               float alpha, float beta) {
    __shared__ float As[TK][16];      // As[kk][cc] = -dis[k0+kk]*A[k0+kk][c0+cc]*dis[c0+cc]
    __shared__ float Hs[TK][BCOL];
    const int c0   = blockIdx.x * 16;
    const int tid  = threadIdx.x;
    const int wave = tid >> 5;
    const int lane = tid & 31;
    const int jcol = wave * 16;
    const int cc   = lane & 15;
    const int kh   = (lane >> 4) * 2;   // K sub-pair select per f32 WMMA layout

    v8f acc = {0.f, 0.f, 0.f, 0.f, 0.f, 0.f, 0.f, 0.f};

    for (int k0 = 0; k0 < NN; k0 += TK) {
        __syncthreads();
        // stage scaled M^T tile (A rows k, cols c0..c0+15)
        for (int i = tid; i < TK * 16; i += 384) {
            const int kk = i >> 4, c = i & 15;
            As[kk][c] = -dis[k0 + kk] * dis[c0 + c] * A[(size_t)(k0 + kk) * NN + c0 + c];
        }
        // stage H chunk (64 x 192), fully coalesced
        for (int i = tid; i < TK * BCOL; i += 384) {
            const int kk = i / BCOL, jj = i % BCOL;
            Hs[kk][jj] = H[(size_t)(k0 + kk) * BCOL + jj];
        }
        __syncthreads();
#pragma unroll
        for (int ks = 0; ks < TK; ks += 4) {
            v2f a, b;
            a.x = As[ks + kh][cc];            // A frag: row=c, K pair per lane half
            a.y = As[ks + kh + 1][cc];
            b.x = Hs[ks + kh][jcol + cc];     // B frag: col=lane&15, K pair per lane half
            b.y = Hs[ks + kh + 1][jcol + cc];
            acc = __builtin_amdgcn_wmma_f32_16x16x4_f32(false, a, false, b,
                                                        (short)0, acc, false, false);
        }
    }

    // C/D layout: VGPR v -> row v (lanes 0-15) / row v+8 (lanes 16-31), col = lane&15
    const int rbase = c0 + ((lane >> 4) ? 8 : 0);
#pragma unroll
    for (int v = 0; v < 8; ++v) {
        const size_t idx = (size_t)(rbase + v) * BCOL + jcol + cc;
        float val = alpha * acc[v];
        if (beta != 0.f) val += beta * Cin[idx];
        Y[idx] = val;
    }
}

// ---------------- out accumulate: out[b,n,o] (+)= T[n, b*12+:12] @ Wk[:, o]
// mode 0: init; 1: add; 2: add + bias + relu
__global__ __launch_bounds__(256) void cheb_accum(const float* __restrict__ T,
                                                  const float* __restrict__ Wk,
                                                  const float* __restrict__ bcheb,
                                                  float* out, int mode) {
    const int i = blockIdx.x * 256 + threadIdx.x;      // [0, 16*2048*64)
    const int o = i & 63;
    const int n = (i >> 6) & (NN - 1);
    const int b = i >> 17;                             // /(64*2048)
    const float* tp = T + (size_t)n * BCOL + b * T1D;
    float acc = 0.f;
#pragma unroll
    for (int t = 0; t < T1D; ++t) acc += tp[t] * Wk[t * T2D + o];
    if (mode == 0) { out[i] = acc; return; }
    acc += out[i];
    if (mode == 2) acc = fmaxf(acc + bcheb[o], 0.f);
    out[i] = acc;
}

extern "C" void kernel_launch(void* const* d_in, const int* in_sizes, int n_in,
                              void* d_out, int out_size, void* d_ws, size_t ws_size,
                              hipStream_t stream) {
    (void)in_sizes; (void)n_in; (void)out_size; (void)ws_size;
    const float* X     = (const float*)d_in[0];
    const float* A     = (const float*)d_in[1];
    // d_in[2..6] (Vs, bs, W1, W2, W3) feed only dead code in the reference.
    const float* Ve    = (const float*)d_in[7];
    const float* be    = (const float*)d_in[8];
    const float* U1    = (const float*)d_in[9];
    const float* U2    = (const float*)d_in[10];
    const float* U3    = (const float*)d_in[11];
    const float* Wcheb = (const float*)d_in[12];
    const float* bcheb = (const float*)d_in[13];
    float* out = (float*)d_out;

    float* ws  = (float*)d_ws;
    float* q   = ws;                    // 192
    float* r   = ws + 192;              // 192
    float* E   = ws + 384;              // 2304
    float* dis = ws + 2688;             // 2048
    float* T0  = ws + 4736;             // 2048*192
    float* T1b = T0 + (size_t)NN * BCOL;// 2048*192   (total ~3.2 MB)

    qr_kernel  <<<16,   256, 0, stream>>>(X, U1, U2, U3, q, r);
    e_kernel   <<<16,   160, 0, stream>>>(q, r, be, Ve, E);
    deg_kernel <<<NN,   256, 0, stream>>>(A, dis);
    xhat_kernel<<<(NN * BCOL) / 256, 256, 0, stream>>>(X, E, T0);

    cheb_accum <<<(16 * NN * T2D) / 256, 256, 0, stream>>>(T0, Wcheb,        bcheb, out, 0);
    lx_gemm    <<<NN / 16, 384, 0, stream>>>(A, dis, T0, T0,  T1b, 1.f,  0.f);
    cheb_accum <<<(16 * NN * T2D) / 256, 256, 0, stream>>>(T1b, Wcheb + 768, bcheb, out, 1);
    lx_gemm    <<<NN / 16, 384, 0, stream>>>(A, dis, T1b, T0, T0,  2.f, -1.f);   // Y==Cin: alias-safe
    cheb_accum <<<(16 * NN * T2D) / 256, 256, 0, stream>>>(T0, Wcheb + 1536, bcheb, out, 2);
}